// get_loss_55216099558213
// MI455X (gfx1250) — compile-verified
//
#include <hip/hip_runtime.h>
#include <math.h>

typedef __attribute__((ext_vector_type(2))) float v2f;
typedef __attribute__((ext_vector_type(8))) float v8f;

constexpr int kBS   = 8;
constexpr int kHW   = 4096;   // 64*64
constexpr int kStep = 20;
constexpr int kV    = 204;    // 4096 // 20
constexpr int kP    = 400;
constexpr int kPT   = 25;     // p tiles of 16 (N axis / lanes)
constexpr int kQT   = 25;     // q tiles of 16 (M axis / rows)

__global__ __launch_bounds__(256)
void adds_wmma_kernel(const float* __restrict__ pred_r,
                      const float* __restrict__ pred_t,
                      const float* __restrict__ pred_score,
                      const float* __restrict__ gt_r,
                      const float* __restrict__ gt_t,
                      const int*   __restrict__ cls_ids,
                      const float* __restrict__ model_xyz,
                      float* __restrict__ wsAdd,
                      float* __restrict__ wsLoss)
{
    // A-operand tiles for the gt side, pre-scaled so WMMA directly yields
    //   D[q][p] = gn[q] - 2*<gt_q, pred_p>      (C = inline 0)
    // gtA[0][q] = (-2*gx, -2*gy)   -> lanes 0-15  (K0,K1)
    // gtA[1][q] = (-2*gz,  gn  )   -> lanes 16-31 (K2,K3)
    __shared__ float2 gtA[2][kP];

    const int b    = blockIdx.x;
    const int tid  = threadIdx.x;
    const int wave = tid >> 5;
    const int lane = tid & 31;

    // ---- per-block: gt = gt_r @ model + gt_t ; pack scaled A tiles ----
    const float G00 = gt_r[b*9+0], G01 = gt_r[b*9+1], G02 = gt_r[b*9+2];
    const float G10 = gt_r[b*9+3], G11 = gt_r[b*9+4], G12 = gt_r[b*9+5];
    const float G20 = gt_r[b*9+6], G21 = gt_r[b*9+7], G22 = gt_r[b*9+8];
    const float T0  = gt_t[b*3+0], T1  = gt_t[b*3+1], T2  = gt_t[b*3+2];
    const int   mb  = b * 3 * kP;
    for (int p = tid; p < kP; p += 256) {
        float mx = model_xyz[mb + p];
        float my = model_xyz[mb + kP + p];
        float mz = model_xyz[mb + 2*kP + p];
        float gx = fmaf(G00,mx, fmaf(G01,my, fmaf(G02,mz, T0)));
        float gy = fmaf(G10,mx, fmaf(G11,my, fmaf(G12,mz, T1)));
        float gz = fmaf(G20,mx, fmaf(G21,my, fmaf(G22,mz, T2)));
        float gn = gx*gx + gy*gy + gz*gz;
        gtA[0][p] = make_float2(-2.f*gx, -2.f*gy);
        gtA[1][p] = make_float2(-2.f*gz,  gn);
    }
    __syncthreads();

    const int  v      = blockIdx.y * 8 + wave;
    const bool vvalid = (v < kV);
    const int  vc     = vvalid ? v : (kV - 1);   // invalid waves run with dummy v
    const int  pix    = vc * kStep;

    // ---- quaternion -> R, t (wave-uniform) ----
    float q0 = pred_r[(b*4+0)*kHW + pix];
    float q1 = pred_r[(b*4+1)*kHW + pix];
    float q2 = pred_r[(b*4+2)*kHW + pix];
    float q3 = pred_r[(b*4+3)*kHW + pix];
    const float inv = rsqrtf(q0*q0 + q1*q1 + q2*q2 + q3*q3);
    q0 *= inv; q1 *= inv; q2 *= inv; q3 *= inv;
    const float R00 = 1.f - 2.f*(q2*q2 + q3*q3);
    const float R01 = 2.f*q1*q2 - 2.f*q0*q3;
    const float R02 = 2.f*q0*q2 + 2.f*q1*q3;
    const float R10 = 2.f*q1*q2 + 2.f*q3*q0;
    const float R11 = 1.f - 2.f*(q1*q1 + q3*q3);
    const float R12 = -2.f*q0*q1 + 2.f*q2*q3;
    const float R20 = -2.f*q0*q2 + 2.f*q1*q3;
    const float R21 = 2.f*q0*q1 + 2.f*q2*q3;
    const float R22 = 1.f - 2.f*(q1*q1 + q2*q2);
    const float t0 = pred_t[(b*3+0)*kHW + pix];
    const float t1 = pred_t[(b*3+1)*kHW + pix];
    const float t2 = pred_t[(b*3+2)*kHW + pix];

    const int  col  = lane & 15;
    const int  half = lane >> 4;    // 0: K0,K1 | 1: K2,K3
    const bool low  = (lane < 16);

    float addSum = 0.f, addsSum = 0.f;

    for (int pt = 0; pt < kPT; ++pt) {
        const int p = pt*16 + col;
        const float mx = model_xyz[mb + p];
        const float my = model_xyz[mb + kP + p];
        const float mz = model_xyz[mb + 2*kP + p];
        const float px = fmaf(R00,mx, fmaf(R01,my, fmaf(R02,mz, t0)));
        const float py = fmaf(R10,mx, fmaf(R11,my, fmaf(R12,mz, t1)));
        const float pz = fmaf(R20,mx, fmaf(R21,my, fmaf(R22,mz, t2)));
        const float pnv = px*px + py*py + pz*pz;

        // matched-point ADD distance (unscale gt from the -2x packed tiles)
        const float2 ga0 = gtA[0][p];
        const float2 ga1 = gtA[1][p];
        const float dx = px + 0.5f*ga0.x;   // px - gx
        const float dy = py + 0.5f*ga0.y;   // py - gy
        const float dz = pz + 0.5f*ga1.x;   // pz - gz
        const float dmatch = sqrtf(dx*dx + dy*dy + dz*dz);
        addSum += low ? dmatch : 0.f;

        // B tile: pred points on the N axis, K padded with 1 (multiplies gn row)
        v2f Bv;
        Bv[0] = half ? pz  : px;
        Bv[1] = half ? 1.f : py;

        v8f m;
        #pragma unroll
        for (int r = 0; r < 8; ++r) m[r] = 3.0e38f;

        for (int qt = 0; qt < kQT; ++qt) {
            const int q = qt*16 + col;
            const float2 av = gtA[half][q];       // one ds_load_b64
            v2f A; A[0] = av.x; A[1] = av.y;
            v8f Cz = {};                           // inline 0 accumulator
            // D[q_row][p_lane] = gn[q] - 2*<gt_q, pred_p>
            v8f D = __builtin_amdgcn_wmma_f32_16x16x4_f32(
                        false, A, false, Bv, (short)0, Cz, false, false);
            #pragma unroll
            for (int r = 0; r < 8; ++r) m[r] = fminf(m[r], D[r]);
        }

        // min over q: tree over 8 row-regs, then merge lane halves (rows 8-15)
        float mn = fminf(fminf(fminf(m[0], m[1]), fminf(m[2], m[3])),
                         fminf(fminf(m[4], m[5]), fminf(m[6], m[7])));
        mn = fminf(mn, __shfl_xor(mn, 16, 32));
        const float term = sqrtf(fmaxf(pnv + mn, 0.f));   // pn[p] + min_q(gn-2c)
        addsSum += low ? term : 0.f;                      // halves are duplicates
    }

    // wave reduction (high-half lanes carry zeros)
    #pragma unroll
    for (int mask = 16; mask >= 1; mask >>= 1) {
        addSum  += __shfl_xor(addSum,  mask, 32);
        addsSum += __shfl_xor(addsSum, mask, 32);
    }

    if (vvalid && lane == 0) {
        const float addMean  = addSum  * (1.f / kP);
        const float addsMean = addsSum * (1.f / kP);
        const int   cls = cls_ids[b];
        const bool  sym = (cls==12)||(cls==15)||(cls==18)||(cls==19)||(cls==20);
        const float a   = sym ? addsMean : addMean;
        const float ps  = pred_score[b*kHW + pix];
        wsAdd [b*kV + v] = a;
        wsLoss[b*kV + v] = a*ps - 0.01f*logf(ps);
    }
}

__global__ __launch_bounds__(256)
void finalize_kernel(const float* __restrict__ wsAdd,
                     const float* __restrict__ wsLoss,
                     const int*   __restrict__ cls_ids,
                     float* __restrict__ out)
{
    __shared__ float sVal[kBS], sLoss[kBS];
    const int wave = threadIdx.x >> 5;   // one wave per batch element
    const int lane = threadIdx.x & 31;
    float sA = 0.f, sL = 0.f;
    for (int v = lane; v < kV; v += 32) {
        sA += wsAdd [wave*kV + v];
        sL += wsLoss[wave*kV + v];
    }
    #pragma unroll
    for (int mask = 16; mask >= 1; mask >>= 1) {
        sA += __shfl_xor(sA, mask, 32);
        sL += __shfl_xor(sL, mask, 32);
    }
    if (lane == 0) {
        const int  cls   = cls_ids[wave];
        const bool valid = ((cls + 1) >= 1) && ((cls + 1) <= 30);
        sVal [wave] = valid ? sA * (1.f / kV) : 0.f;
        sLoss[wave] = valid ? sL * (1.f / kV) : 0.f;
    }
    __syncthreads();
    if (threadIdx.x == 0) {
        float gl = 0.f, gv = 0.f;
        for (int i = 0; i < kBS; ++i) { gl += sLoss[i]; gv += sVal[i]; }
        gl *= (1.f / kBS);
        gv *= (1.f / kBS);
        if (isnan(gl) || isinf(gl)) gl = 0.f;
        out[0] = gl + 0.f * gv;
    }
}

extern "C" void kernel_launch(void* const* d_in, const int* in_sizes, int n_in,
                              void* d_out, int out_size, void* d_ws, size_t ws_size,
                              hipStream_t stream) {
    const float* pred_r     = (const float*)d_in[0];
    const float* pred_t     = (const float*)d_in[1];
    const float* pred_score = (const float*)d_in[2];
    const float* gt_r       = (const float*)d_in[3];
    const float* gt_t       = (const float*)d_in[4];
    const int*   cls_ids    = (const int*)  d_in[5];
    const float* model_xyz  = (const float*)d_in[6];
    float* out    = (float*)d_out;
    float* wsAdd  = (float*)d_ws;
    float* wsLoss = wsAdd + kBS * kV;

    dim3 grid(kBS, (kV + 7) / 8);
    adds_wmma_kernel<<<grid, 256, 0, stream>>>(pred_r, pred_t, pred_score,
                                               gt_r, gt_t, cls_ids, model_xyz,
                                               wsAdd, wsLoss);
    finalize_kernel<<<1, 256, 0, stream>>>(wsAdd, wsLoss, cls_ids, out);
}